// Attention_9981503996367
// MI455X (gfx1250) — compile-verified
//
#include <hip/hip_runtime.h>
#include <hip/hip_bf16.h>
#include <math.h>

// ---------------------------------------------------------------------------
// MI455X (gfx1250) fp32 causal multi-head attention.
// All GEMM-like math runs on V_WMMA_F32_16X16X4_F32 (wave32 matrix core).
//
// Fragment layouts used (CDNA5 ISA 7.12.2, f32 16x16x4):
//   A (16x4, MxK): lane L, elem v -> A[L%16][(L/16)*2 + v]
//   B (4x16, KxN): lane L, elem v -> B[(L/16)*2 + v][L%16]
//   C/D (16x16):   VGPR r, lane L -> C[r + (L/16)*8][L%16]
// ---------------------------------------------------------------------------

typedef __attribute__((ext_vector_type(2))) float v2f;
typedef __attribute__((ext_vector_type(8))) float v8f;

#define DMODEL 1024
#define NHEADS 16
#define HDIM   64
#define BATCH  4
#define SEQ    2048
#define MROWS  (BATCH * SEQ)          // 8192
#define QSZ    ((size_t)MROWS * DMODEL) // floats per intermediate buffer

static __device__ __forceinline__ v8f wmma4(v2f a, v2f b, v8f c) {
    return __builtin_amdgcn_wmma_f32_16x16x4_f32(false, a, false, b,
                                                 (short)0, c, false, false);
}

// ---------------------------------------------------------------------------
// Kernel 1: fused Q/K/V projection.  y = x @ W + b, stored head-split as
// [B,H,T,Hd].  One wave = one 16(M) x 64(N) strip; task space covers 3 GEMMs.
// ---------------------------------------------------------------------------
__global__ __launch_bounds__(256)
void qkv_proj_kernel(const float* __restrict__ x,
                     const float* __restrict__ Wq, const float* __restrict__ bq,
                     const float* __restrict__ Wk, const float* __restrict__ bk,
                     const float* __restrict__ Wv, const float* __restrict__ bv,
                     float* __restrict__ Q, float* __restrict__ K,
                     float* __restrict__ V) {
    const int lane = threadIdx.x & 31;
    const int wave = threadIdx.x >> 5;
    const int half = lane >> 4;     // 0: lanes 0-15, 1: lanes 16-31
    const int l16  = lane & 15;

    const int tasks_per_gemm = (MROWS / 16) * (DMODEL / 64);   // 512*16 = 8192
    int task = blockIdx.x * 8 + wave;
    const int g    = task / tasks_per_gemm;      // 0:Q 1:K 2:V
    int rem        = task - g * tasks_per_gemm;
    const int mt   = rem >> 4;                   // 0..511
    const int ng   = rem & 15;                   // 0..15  (head index)
    const int m0   = mt * 16;
    const int n0   = ng * 64;

    const float* __restrict__ W    = (g == 0) ? Wq : (g == 1) ? Wk : Wv;
    const float* __restrict__ bias = (g == 0) ? bq : (g == 1) ? bk : bv;
    float* __restrict__ out        = (g == 0) ? Q  : (g == 1) ? K  : V;

    v8f acc0 = {}, acc1 = {}, acc2 = {}, acc3 = {};

    const float* __restrict__ arow = x + (size_t)(m0 + l16) * DMODEL + half * 2;
    const float* __restrict__ wp   = W + n0 + l16;

    for (int k0 = 0; k0 < DMODEL; k0 += 4) {
        v2f a;
        a.x = arow[k0];
        a.y = arow[k0 + 1];
        const size_t wk = (size_t)(k0 + half * 2) * DMODEL;
        v2f b0, b1, b2, b3;
        b0.x = wp[wk +  0];           b0.y = wp[wk + DMODEL +  0];
        b1.x = wp[wk + 16];           b1.y = wp[wk + DMODEL + 16];
        b2.x = wp[wk + 32];           b2.y = wp[wk + DMODEL + 32];
        b3.x = wp[wk + 48];           b3.y = wp[wk + DMODEL + 48];
        acc0 = wmma4(a, b0, acc0);
        acc1 = wmma4(a, b1, acc1);
        acc2 = wmma4(a, b2, acc2);
        acc3 = wmma4(a, b3, acc3);
    }

    const float bb0 = bias[n0 +  0 + l16];
    const float bb1 = bias[n0 + 16 + l16];
    const float bb2 = bias[n0 + 32 + l16];
    const float bb3 = bias[n0 + 48 + l16];

    const int h = ng;                // n0/64
#pragma unroll
    for (int r = 0; r < 8; ++r) {
        const int m  = m0 + r + half * 8;     // global row in [0,8192)
        const int b  = m >> 11;               // /SEQ
        const int tt = m & (SEQ - 1);
        float* o = out + (((size_t)b * NHEADS + h) * SEQ + tt) * HDIM;
        o[ 0 + l16] = acc0[r] + bb0;
        o[16 + l16] = acc1[r] + bb1;
        o[32 + l16] = acc2[r] + bb2;
        o[48 + l16] = acc3[r] + bb3;
    }
}

// ---------------------------------------------------------------------------
// Kernel 2: flash-style causal attention.  One wave = one (b,h) head and one
// 16-row query tile.  Online softmax; S and O products on fp32 WMMA.
// ---------------------------------------------------------------------------
__global__ __launch_bounds__(256)
void attn_kernel(const float* __restrict__ Q, const float* __restrict__ K,
                 const float* __restrict__ V, float* __restrict__ ctx) {
    __shared__ float lds[8][16][17];          // per-wave P bounce tile (padded)

    const int lane = threadIdx.x & 31;
    const int wave = threadIdx.x >> 5;
    const int half = lane >> 4;
    const int l16  = lane & 15;

    const int task = blockIdx.x * 8 + wave;   // 0..8191
    const int qt   = task & 127;              // query tile index (T/16)
    const int bh   = task >> 7;               // b*H + h
    const int q0   = qt * 16;

    const size_t base = (size_t)bh * SEQ * HDIM;
    const float* __restrict__ Qp = Q + base;
    const float* __restrict__ Kp = K + base;
    const float* __restrict__ Vp = V + base;

    // Q fragments held in registers: qa[kk] covers head-dims kk*4 .. kk*4+3
    v2f qa[16];
    {
        const float* qrow = Qp + (size_t)(q0 + l16) * HDIM + half * 2;
#pragma unroll
        for (int kk = 0; kk < 16; ++kk) {
            qa[kk].x = qrow[kk * 4];
            qa[kk].y = qrow[kk * 4 + 1];
        }
    }

    v8f o0 = {}, o1 = {}, o2 = {}, o3 = {};
    float mrow[8], lrow[8];
#pragma unroll
    for (int r = 0; r < 8; ++r) { mrow[r] = -__builtin_inff(); lrow[r] = 0.f; }

    float (*tile)[17] = lds[wave];
    const float scale = 0.125f;               // 1/sqrt(64)

    for (int j = 0; j <= qt; ++j) {
        const int k0 = j * 16;

        if (j + 1 <= qt) {                    // prefetch next K tile row
            __builtin_prefetch(Kp + (size_t)(k0 + 16 + l16) * HDIM, 0, 3);
        }

        // ---- S = Q @ K^T  (K-dim = head dim, 64 = 16 WMMA steps) ----
        v8f s = {};
        const float* krow = Kp + (size_t)(k0 + l16) * HDIM + half * 2;
#pragma unroll
        for (int kk = 0; kk < 16; ++kk) {
            v2f bf;
            bf.x = krow[kk * 4];
            bf.y = krow[kk * 4 + 1];
            s = wmma4(qa[kk], bf, s);
        }

        // ---- mask + online softmax (rows live in 16-lane half-waves) ----
#pragma unroll
        for (int r = 0; r < 8; ++r) {
            const int M = r + half * 8;       // row within tile
            float sv = s[r] * scale;
            if (j == qt && l16 > M) sv = -__builtin_inff();
            float rm = sv;
            rm = fmaxf(rm, __shfl_xor(rm, 1, 32));
            rm = fmaxf(rm, __shfl_xor(rm, 2, 32));
            rm = fmaxf(rm, __shfl_xor(rm, 4, 32));
            rm = fmaxf(rm, __shfl_xor(rm, 8, 32));
            const float mnew = fmaxf(mrow[r], rm);
            const float p    = __expf(sv - mnew);
            float rs = p;
            rs += __shfl_xor(rs, 1, 32);
            rs += __shfl_xor(rs, 2, 32);
            rs += __shfl_xor(rs, 4, 32);
            rs += __shfl_xor(rs, 8, 32);
            const float corr = __expf(mrow[r] - mnew);
            lrow[r] = lrow[r] * corr + rs;
            mrow[r] = mnew;
            s[r]    = p;
            o0[r] *= corr; o1[r] *= corr; o2[r] *= corr; o3[r] *= corr;
        }

        // ---- C-layout -> A-layout for P via per-wave LDS bounce ----
#pragma unroll
        for (int r = 0; r < 8; ++r)
            tile[r + half * 8][l16] = s[r];
        asm volatile("s_wait_dscnt 0x0" ::: "memory");
        v2f pa[4];
#pragma unroll
        for (int kk = 0; kk < 4; ++kk) {
            pa[kk].x = tile[l16][kk * 4 + half * 2];
            pa[kk].y = tile[l16][kk * 4 + half * 2 + 1];
        }
        asm volatile("s_wait_dscnt 0x0" ::: "memory");

        // ---- O += P @ V  (K-dim = 16 = 4 WMMA steps, 4 Hd chunks) ----
#pragma unroll
        for (int kk = 0; kk < 4; ++kk) {
            const float* vrow = Vp + (size_t)(k0 + kk * 4 + half * 2) * HDIM;
            v2f vb0, vb1, vb2, vb3;
            vb0.x = vrow[ 0 + l16]; vb0.y = vrow[HDIM +  0 + l16];
            vb1.x = vrow[16 + l16]; vb1.y = vrow[HDIM + 16 + l16];
            vb2.x = vrow[32 + l16]; vb2.y = vrow[HDIM + 32 + l16];
            vb3.x = vrow[48 + l16]; vb3.y = vrow[HDIM + 48 + l16];
            o0 = wmma4(pa[kk], vb0, o0);
            o1 = wmma4(pa[kk], vb1, o1);
            o2 = wmma4(pa[kk], vb2, o2);
            o3 = wmma4(pa[kk], vb3, o3);
        }
    }

    // ---- normalize and store ctx in [B,T,D] layout ----
    const int h_ = bh & (NHEADS - 1);
    const int b_ = bh >> 4;
#pragma unroll
    for (int r = 0; r < 8; ++r) {
        const float inv = 1.0f / lrow[r];
        const int tq = q0 + r + half * 8;
        float* o = ctx + ((size_t)b_ * SEQ + tq) * DMODEL + h_ * HDIM;
        o[ 0 + l16] = o0[r] * inv;
        o[16 + l16] = o1[r] * inv;
        o[32 + l16] = o2[r] * inv;
        o[48 + l16] = o3[r] * inv;
    }
}

// ---------------------------------------------------------------------------
// Kernel 3: output projection.  out = ctx @ Wo + bo, plain row-major.
// ---------------------------------------------------------------------------
__global__ __launch_bounds__(256)
void out_proj_kernel(const float* __restrict__ A, const float* __restrict__ W,
                     const float* __restrict__ bias, float* __restrict__ out) {
    const int lane = threadIdx.x & 31;
    const int wave = threadIdx.x >> 5;
    const int half = lane >> 4;
    const int l16  = lane & 15;

    int task = blockIdx.x * 8 + wave;        // 8192 tasks
    const int mt = task >> 4;
    const int ng = task & 15;
    const int m0 = mt * 16;
    const int n0 = ng * 64;

    v8f acc0 = {}, acc1 = {}, acc2 = {}, acc3 = {};

    const float* __restrict__ arow = A + (size_t)(m0 + l16) * DMODEL + half * 2;
    const float* __restrict__ wp   = W + n0 + l16;

    for (int k0 = 0; k0 < DMODEL; k0 += 4) {
        v2f a;
        a.x = arow[k0];
        a.y = arow[k0 + 1];
        const size_t wk = (size_t)(k0 + half * 2) * DMODEL;
        v2f b0, b1, b2, b3;
        b0.x = wp[wk +  0];           b0.y = wp[wk + DMODEL +  0];
        b1.x = wp[wk + 16];           b1.y = wp[wk + DMODEL + 16];
        b2.x = wp[wk + 32];           b2.y = wp[wk + DMODEL + 32];
        b3.x = wp[wk + 48];           b3.y = wp[wk + DMODEL + 48];
        acc0 = wmma4(a, b0, acc0);
        acc1 = wmma4(a, b1, acc1);
        acc2 = wmma4(a, b2, acc2);
        acc3 = wmma4(a, b3, acc3);
    }

    const float bb0 = bias[n0 +  0 + l16];
    const float bb1 = bias[n0 + 16 + l16];
    const float bb2 = bias[n0 + 32 + l16];
    const float bb3 = bias[n0 + 48 + l16];

#pragma unroll
    for (int r = 0; r < 8; ++r) {
        const int m = m0 + r + half * 8;
        float* o = out + (size_t)m * DMODEL + n0;
        o[ 0 + l16] = acc0[r] + bb0;
        o[16 + l16] = acc1[r] + bb1;
        o[32 + l16] = acc2[r] + bb2;
        o[48 + l16] = acc3[r] + bb3;
    }
}

// ---------------------------------------------------------------------------
extern "C" void kernel_launch(void* const* d_in, const int* in_sizes, int n_in,
                              void* d_out, int out_size, void* d_ws,
                              size_t ws_size, hipStream_t stream) {
    const float* x  = (const float*)d_in[0];
    const float* Wq = (const float*)d_in[1];
    const float* bq = (const float*)d_in[2];
    const float* Wk = (const float*)d_in[3];
    const float* bk = (const float*)d_in[4];
    const float* Wv = (const float*)d_in[5];
    const float* bv = (const float*)d_in[6];
    const float* Wo = (const float*)d_in[7];
    const float* bo = (const float*)d_in[8];
    float* out = (float*)d_out;

    float* Qb = (float*)d_ws;
    float* Kb = Qb + QSZ;
    float* Vb = Kb + QSZ;
    float* Cb = Vb + QSZ;

    // 3 GEMMs x 8192 wave-tasks / 8 waves per block
    qkv_proj_kernel<<<3072, 256, 0, stream>>>(x, Wq, bq, Wk, bk, Wv, bv,
                                              Qb, Kb, Vb);
    // 4*16*128 = 8192 wave-tasks / 8
    attn_kernel<<<1024, 256, 0, stream>>>(Qb, Kb, Vb, Cb);
    // 8192 wave-tasks / 8
    out_proj_kernel<<<1024, 256, 0, stream>>>(Cb, Wo, bo, out);
}